// GAT_34153579937840
// MI455X (gfx1250) — compile-verified
//
#include <hip/hip_runtime.h>
#include <cstdint>
#include <cstddef>

#define NNODES 50000
#define NEDGES 800000
#define DIN0   128
#define DHID   64
#define NEG_SLOPE 0.2f

typedef __attribute__((ext_vector_type(2))) float v2f;
typedef __attribute__((ext_vector_type(8))) float v8f;

// ---- monotone float<->uint encoding so u32 atomicMax == float max ----
__device__ __forceinline__ unsigned encf(float f) {
  unsigned u = __float_as_uint(f);
  return (u & 0x80000000u) ? ~u : (u | 0x80000000u);
}
__device__ __forceinline__ float decf(unsigned u) {
  unsigned v = (u & 0x80000000u) ? (u & 0x7FFFFFFFu) : ~u;
  return __uint_as_float(v);
}

// ------------------------------- utility fills -------------------------------
__global__ void k_fill_f32(float* __restrict__ p, float v, int n) {
  int i = blockIdx.x * blockDim.x + threadIdx.x;
  if (i < n) p[i] = v;
}
__global__ void k_fill_u32(unsigned* __restrict__ p, unsigned v, int n) {
  int i = blockIdx.x * blockDim.x + threadIdx.x;
  if (i < n) p[i] = v;
}

// ------------------- one-time: in-degree + weighted sum per dst --------------
__global__ void k_degsum(const int* __restrict__ dst, const float* __restrict__ w,
                         float* __restrict__ deg, float* __restrict__ wsum) {
  int e = blockIdx.x * blockDim.x + threadIdx.x;
  if (e >= NEDGES) return;
  int d = dst[e];
  atomicAdd(deg + d, 1.0f);
  atomicAdd(wsum + d, w[e]);
}
__global__ void k_loopattr(const float* __restrict__ deg, const float* __restrict__ wsum,
                           float* __restrict__ la) {
  int i = blockIdx.x * blockDim.x + threadIdx.x;
  if (i < NNODES) la[i] = wsum[i] / fmaxf(deg[i], 1.0f);
}

// ---------------- per-layer scalar: ce = sum_j W_e[0,j] * a_e[j] -------------
__global__ void k_ce(const float* __restrict__ We, const float* __restrict__ ae,
                     float* __restrict__ ce, int dout) {
  if (blockIdx.x == 0 && threadIdx.x == 0) {
    float s = 0.f;
    for (int j = 0; j < dout; ++j) s += We[j] * ae[j];
    *ce = s;
  }
}

// --------------------- H = X @ W via V_WMMA_F32_16X16X4_F32 ------------------
// grid.x = N/16 row tiles; blockDim = (32, dout/16): one wave per 16x16 tile.
__global__ void k_gemm_wmma(const float* __restrict__ X, const float* __restrict__ W,
                            float* __restrict__ H, int din, int dout) {
  int row0 = blockIdx.x * 16;
  int col0 = threadIdx.y * 16;
  int lane = threadIdx.x;
  int half = lane >> 4;     // 0: K pair {0,1}, 1: K pair {2,3}
  int lr   = lane & 15;
  v8f c = {};
  const float* xrow = X + (size_t)(row0 + lr) * din;
  for (int k0 = 0; k0 < din; k0 += 4) {
    int kb = k0 + 2 * half;
    v2f a, b;
    a.x = xrow[kb];
    a.y = xrow[kb + 1];
    b.x = W[(size_t)kb * dout + col0 + lr];
    b.y = W[(size_t)(kb + 1) * dout + col0 + lr];
    c = __builtin_amdgcn_wmma_f32_16x16x4_f32(false, a, false, b, (short)0, c,
                                              false, false);
  }
#pragma unroll
  for (int r = 0; r < 8; ++r) {
    int row = row0 + r + 8 * half;   // C/D layout: VGPR r holds M=r / M=r+8
    H[(size_t)row * dout + col0 + lr] = c[r];
  }
}

// last layer (dout == 1): plain per-node dot, float4 loads
__global__ void k_gemv1(const float* __restrict__ X, const float* __restrict__ W,
                        float* __restrict__ H, int din) {
  int i = blockIdx.x * blockDim.x + threadIdx.x;
  if (i >= NNODES) return;
  const float4* xr = (const float4*)(X + (size_t)i * din);
  const float4* w4 = (const float4*)W;
  float s = 0.f;
  for (int k = 0; k < din / 4; ++k) {
    float4 xv = xr[k], wv = w4[k];
    s += xv.x * wv.x + xv.y * wv.y + xv.z * wv.z + xv.w * wv.w;
  }
  H[i] = s;
}

// ------------------- per-node attention dots: h·a_s, h·a_d -------------------
__global__ void k_alphadot(const float* __restrict__ H, const float* __restrict__ aw_s,
                           const float* __restrict__ aw_d, float* __restrict__ as_,
                           float* __restrict__ ad_, int dout) {
  int i = blockIdx.x * blockDim.x + threadIdx.x;
  if (i >= NNODES) return;
  const float* h = H + (size_t)i * dout;
  float s = 0.f, d = 0.f;
  int f = 0;
  for (; f + 4 <= dout; f += 4) {
    float4 hv = *(const float4*)(h + f);
    float4 sv = *(const float4*)(aw_s + f);
    float4 dv = *(const float4*)(aw_d + f);
    s += hv.x * sv.x + hv.y * sv.y + hv.z * sv.z + hv.w * sv.w;
    d += hv.x * dv.x + hv.y * dv.y + hv.z * dv.z + hv.w * dv.w;
  }
  for (; f < dout; ++f) {           // tail (covers dout == 1)
    s += h[f] * aw_s[f];
    d += h[f] * aw_d[f];
  }
  as_[i] = s;
  ad_[i] = d;
}

// ------- pass 1: alpha = leaky_relu(...), segment max via encoded u32 --------
__global__ void k_edge_alpha(const int* __restrict__ src, const int* __restrict__ dst,
                             const float* __restrict__ w, const float* __restrict__ la,
                             const float* __restrict__ as_, const float* __restrict__ ad_,
                             const float* __restrict__ cep, float* __restrict__ alpha,
                             unsigned* __restrict__ amax) {
  int e = blockIdx.x * blockDim.x + threadIdx.x;
  const int tot = NEDGES + NNODES;
  if (e >= tot) return;
  int s, d; float attr;
  if (e < NEDGES) { s = src[e]; d = dst[e]; attr = w[e]; }
  else            { s = d = e - NEDGES;     attr = la[e - NEDGES]; }
  float al = as_[s] + ad_[d] + (*cep) * attr;
  al = (al > 0.f) ? al : NEG_SLOPE * al;
  alpha[e] = al;
  atomicMax(amax + d, encf(al));
}

// ----------- pass 2: ex = exp(alpha - amax[dst]), segment sum ----------------
__global__ void k_edge_exp(const int* __restrict__ dst, float* __restrict__ alpha,
                           const unsigned* __restrict__ amax, float* __restrict__ denom) {
  int e = blockIdx.x * blockDim.x + threadIdx.x;
  const int tot = NEDGES + NNODES;
  if (e >= tot) return;
  int d = (e < NEDGES) ? dst[e] : (e - NEDGES);
  float ex = __expf(alpha[e] - decf(amax[d]));
  alpha[e] = ex;                 // alpha buffer reused as ex buffer
  atomicAdd(denom + d, ex);
}

// ------- pass 3 (dout=64): out[dst,f] += h[src,f]*coef, 4 feats/thread -------
__global__ void k_edge_agg64(const int* __restrict__ src, const int* __restrict__ dst,
                             const float* __restrict__ ex, const float* __restrict__ denom,
                             const float* __restrict__ H, float* __restrict__ out) {
  long long gid = (long long)blockIdx.x * blockDim.x + threadIdx.x;
  const long long total = (long long)(NEDGES + NNODES) << 4;   // 16 threads / edge
  if (gid >= total) return;
  int e  = (int)(gid >> 4);
  int f0 = ((int)gid & 15) << 2;
  int s, d;
  if (e < NEDGES) { s = src[e]; d = dst[e]; }
  else            { s = d = e - NEDGES; }
  float coef = ex[e] / denom[d];
  float4 hv = *(const float4*)(H + ((size_t)s << 6) + f0);   // global_load_b128
  float* o = out + ((size_t)d << 6) + f0;
  atomicAdd(o + 0, hv.x * coef);
  atomicAdd(o + 1, hv.y * coef);
  atomicAdd(o + 2, hv.z * coef);
  atomicAdd(o + 3, hv.w * coef);
}

// ------- pass 3 (dout=1): scalar aggregation --------------------------------
__global__ void k_edge_agg1(const int* __restrict__ src, const int* __restrict__ dst,
                            const float* __restrict__ ex, const float* __restrict__ denom,
                            const float* __restrict__ H, float* __restrict__ out) {
  int e = blockIdx.x * blockDim.x + threadIdx.x;
  const int tot = NEDGES + NNODES;
  if (e >= tot) return;
  int s, d;
  if (e < NEDGES) { s = src[e]; d = dst[e]; }
  else            { s = d = e - NEDGES; }
  float coef = ex[e] / denom[d];
  atomicAdd(out + d, H[s] * coef);
}

// ------------------------ bias + optional ReLU, in place ---------------------
__global__ void k_finalize(float* __restrict__ buf, const float* __restrict__ bias,
                           int dout, int do_relu) {
  int gid = blockIdx.x * blockDim.x + threadIdx.x;
  if (gid >= NNODES * dout) return;
  float v = buf[gid] + bias[gid & (dout - 1)];   // dout is a power of two (64 or 1)
  if (do_relu) v = fmaxf(v, 0.f);
  buf[gid] = v;
}

// -----------------------------------------------------------------------------
extern "C" void kernel_launch(void* const* d_in, const int* in_sizes, int n_in,
                              void* d_out, int out_size, void* d_ws, size_t ws_size,
                              hipStream_t stream) {
  (void)in_sizes; (void)n_in; (void)out_size; (void)ws_size;
  const float* x   = (const float*)d_in[0];
  const int*   ei  = (const int*)d_in[1];
  const float* ew  = (const float*)d_in[2];
  const int* src = ei;
  const int* dst = ei + NEDGES;

  // ---- carve workspace (256B aligned) ----
  uintptr_t base = (uintptr_t)d_ws;
  size_t off = 0;
  auto carve = [&](size_t bytes) -> void* {
    off = (off + 255) & ~(size_t)255;
    void* p = (void*)(base + off);
    off += bytes;
    return p;
  };
  float*    h      = (float*)carve((size_t)NNODES * DHID * 4);
  float*    featA  = (float*)carve((size_t)NNODES * DHID * 4);
  float*    featB  = (float*)carve((size_t)NNODES * DHID * 4);
  float*    exbuf  = (float*)carve((size_t)(NEDGES + NNODES) * 4);
  float*    as_    = (float*)carve((size_t)NNODES * 4);
  float*    ad_    = (float*)carve((size_t)NNODES * 4);
  float*    denom  = (float*)carve((size_t)NNODES * 4);
  unsigned* amax   = (unsigned*)carve((size_t)NNODES * 4);
  float*    deg    = (float*)carve((size_t)NNODES * 4);
  float*    wsum   = (float*)carve((size_t)NNODES * 4);
  float*    lattr  = (float*)carve((size_t)NNODES * 4);
  float*    ce     = (float*)carve(256);

  const int TB = 256;
  const int tot = NEDGES + NNODES;
  auto blk = [&](long long n) { return (unsigned)((n + TB - 1) / TB); };

  // ---- one-time: self-loop edge attr = mean of incoming edge weights ----
  k_fill_f32<<<blk(NNODES), TB, 0, stream>>>(deg, 0.f, NNODES);
  k_fill_f32<<<blk(NNODES), TB, 0, stream>>>(wsum, 0.f, NNODES);
  k_degsum<<<blk(NEDGES), TB, 0, stream>>>(dst, ew, deg, wsum);
  k_loopattr<<<blk(NNODES), TB, 0, stream>>>(deg, wsum, lattr);

  const float* in = x;
  float* feat[2] = {featA, featB};

  for (int l = 0; l < 4; ++l) {
    int din  = (l == 0) ? DIN0 : DHID;
    int dout = (l == 3) ? 1 : DHID;
    const float* W    = (const float*)d_in[3 + 6 * l + 0];
    const float* a_s  = (const float*)d_in[3 + 6 * l + 1];
    const float* a_d  = (const float*)d_in[3 + 6 * l + 2];
    const float* W_e  = (const float*)d_in[3 + 6 * l + 3];
    const float* a_e  = (const float*)d_in[3 + 6 * l + 4];
    const float* bias = (const float*)d_in[3 + 6 * l + 5];
    float* agg = (l == 3) ? (float*)d_out : feat[l & 1];

    // init per-layer accumulators
    k_fill_f32<<<blk((long long)NNODES * dout), TB, 0, stream>>>(agg, 0.f, NNODES * dout);
    k_fill_f32<<<blk(NNODES), TB, 0, stream>>>(denom, 0.f, NNODES);
    k_fill_u32<<<blk(NNODES), TB, 0, stream>>>(amax, 0u, NNODES);  // 0u == -inf in encoding
    k_ce<<<1, 32, 0, stream>>>(W_e, a_e, ce, dout);

    // h = in @ W
    if (dout == DHID) {
      dim3 b(32, DHID / 16);                    // 4 waves: one per 16-col tile
      k_gemm_wmma<<<dim3(NNODES / 16), b, 0, stream>>>(in, W, h, din, dout);
    } else {
      k_gemv1<<<blk(NNODES), TB, 0, stream>>>(in, W, h, din);
    }

    // per-node attention dots
    k_alphadot<<<blk(NNODES), TB, 0, stream>>>(h, a_s, a_d, as_, ad_, dout);

    // edge passes: leaky-relu + segment max, exp + segment sum, aggregation
    k_edge_alpha<<<blk(tot), TB, 0, stream>>>(src, dst, ew, lattr, as_, ad_, ce,
                                              exbuf, amax);
    k_edge_exp<<<blk(tot), TB, 0, stream>>>(dst, exbuf, amax, denom);
    if (dout == DHID) {
      k_edge_agg64<<<blk((long long)tot << 4), TB, 0, stream>>>(src, dst, exbuf,
                                                                denom, h, agg);
    } else {
      k_edge_agg1<<<blk(tot), TB, 0, stream>>>(src, dst, exbuf, denom, h, agg);
    }

    // bias (+ ReLU for hidden layers), in place
    k_finalize<<<blk((long long)NNODES * dout), TB, 0, stream>>>(agg, bias, dout,
                                                                 (l < 3) ? 1 : 0);
    in = agg;
  }
}